// SphericalHarmonics_1151051235358
// MI455X (gfx1250) — compile-verified
//
#include <hip/hip_runtime.h>
#include <math.h>

typedef __attribute__((ext_vector_type(2))) float v2f;
typedef __attribute__((ext_vector_type(8))) float v8f;

#define S3   1.7320508075688772f
#define S5   2.2360679774997896f
#define S7   2.6457513110645907f
#define S15  3.8729833462074170f
#define S42  6.4807406984078600f
#define S70  8.3666002653407560f
#define S105 10.246950765959598f

// B[20][16]: SH(l<=3) coefficients over the 20 monomials
// m: 0=1 1=x 2=y 3=z 4=x2 5=y2 6=z2 7=xy 8=xz 9=yz
//    10=x3 11=y3 12=z3 13=x2y 14=x2z 15=xy2 16=y2z 17=xz2 18=yz2 19=xyz
__device__ __forceinline__ float bcoef(int k, int n) {
  float v = 0.0f;                                     // branch-free: pure cndmask chain
  v = (n==0  && k==0 ) ?  1.0f        : v;            // 1
  v = (n==1  && k==2 ) ?  S3          : v;            // s3*y
  v = (n==2  && k==3 ) ?  S3          : v;            // s3*z
  v = (n==3  && k==1 ) ?  S3          : v;            // s3*x
  v = (n==4  && k==7 ) ?  S15         : v;            // s15*xy
  v = (n==5  && k==9 ) ?  S15         : v;            // s15*yz
  v = (n==6  && k==6 ) ?  1.5f*S5     : v;            // .5*s5*(3z2-1)
  v = (n==6  && k==0 ) ? -0.5f*S5     : v;
  v = (n==7  && k==8 ) ?  S15         : v;            // s15*xz
  v = (n==8  && k==4 ) ?  0.5f*S15    : v;            // .5*s15*(x2-y2)
  v = (n==8  && k==5 ) ? -0.5f*S15    : v;
  v = (n==9  && k==13) ?  0.75f*S70   : v;            // .25*s70*y*(3x2-y2)
  v = (n==9  && k==11) ? -0.25f*S70   : v;
  v = (n==10 && k==19) ?  S105        : v;            // s105*xyz
  v = (n==11 && k==18) ?  1.25f*S42   : v;            // .25*s42*y*(5z2-1)
  v = (n==11 && k==2 ) ? -0.25f*S42   : v;
  v = (n==12 && k==12) ?  2.5f*S7     : v;            // .5*s7*z*(5z2-3)
  v = (n==12 && k==3 ) ? -1.5f*S7     : v;
  v = (n==13 && k==17) ?  1.25f*S42   : v;            // .25*s42*x*(5z2-1)
  v = (n==13 && k==1 ) ? -0.25f*S42   : v;
  v = (n==14 && k==14) ?  0.5f*S105   : v;            // .5*s105*z*(x2-y2)
  v = (n==14 && k==16) ? -0.5f*S105   : v;
  v = (n==15 && k==10) ?  0.25f*S70   : v;            // .25*s70*x*(x2-3y2)
  v = (n==15 && k==15) ? -0.75f*S70   : v;
  return v;
}

// One wave computes a 16-point x 16-basis tile: D(16x16) = M(16x20) x B(20x16)
// via 5 chained V_WMMA_F32_16X16X4_F32.
__global__ void sph_wmma_kernel(const float* __restrict__ in,
                                float* __restrict__ out,
                                long nTiles) {
  const int lane = (int)(threadIdx.x & 31u);
  const int n    = lane & 15;     // output column / point-in-tile
  const int h    = lane >> 4;     // half-wave: K-pair select, row+8 select
  const long wavesPerBlock = (long)(blockDim.x >> 5);
  long       wave   = (long)blockIdx.x * wavesPerBlock + (long)(threadIdx.x >> 5);
  const long nWaves = (long)gridDim.x * wavesPerBlock;

  // Lane-resident constant B fragments (4x16 f32 each), built once.
  v2f b[5];
#pragma unroll
  for (int s = 0; s < 5; ++s) {
    b[s].x = bcoef(4 * s + 2 * h + 0, n);
    b[s].y = bcoef(4 * s + 2 * h + 1, n);
  }

  for (long tile = wave; tile < nTiles; tile += nWaves) {
    // Both half-waves read the same 16 points (L0/L2 dedups; HBM sees it once).
    const long p  = tile * 16 + n;
    const float vx = in[3 * p + 0];
    const float vy = in[3 * p + 1];
    const float vz = in[3 * p + 2];

    const long nt = tile + nWaves;                  // wave-uniform condition
    if (nt < nTiles) __builtin_prefetch(in + nt * 48 + 2 * lane, 0, 0);

    const float s2  = vx * vx + vy * vy + vz * vz;
    const float inv = 1.0f / fmaxf(sqrtf(s2), 1e-12f);
    const float x = vx * inv, y = vy * inv, z = vz * inv;
    const float x2 = x * x, y2 = y * y, z2 = z * z;

    float m[20];
    m[0]  = 1.0f;    m[1]  = x;       m[2]  = y;       m[3]  = z;
    m[4]  = x2;      m[5]  = y2;      m[6]  = z2;
    m[7]  = x * y;   m[8]  = x * z;   m[9]  = y * z;
    m[10] = x * x2;  m[11] = y * y2;  m[12] = z * z2;
    m[13] = x2 * y;  m[14] = x2 * z;  m[15] = x * y2;
    m[16] = y2 * z;  m[17] = x * z2;  m[18] = y * z2;
    m[19] = m[7] * z;

    v8f c = {0.f, 0.f, 0.f, 0.f, 0.f, 0.f, 0.f, 0.f};
#pragma unroll
    for (int s = 0; s < 5; ++s) {
      v2f a;                                          // A 16x4 f32 fragment
      a.x = h ? m[4 * s + 2] : m[4 * s + 0];
      a.y = h ? m[4 * s + 3] : m[4 * s + 1];
      c = __builtin_amdgcn_wmma_f32_16x16x4_f32(
          /*neg_a=*/false, a, /*neg_b=*/false, b[s],
          /*c_mod=*/(short)0, c, /*reuse_a=*/false, /*reuse_b=*/false);
    }

    // D layout: VGPR r -> row (r + 8h), col n. Streaming NT stores.
    float* o = out + (((tile * 16) + (long)(h * 8)) << 4) + n;
#pragma unroll
    for (int r = 0; r < 8; ++r) {
      __builtin_nontemporal_store(c[r], o + ((long)r << 4));
    }
  }
}

// Scalar tail for npts % 16 (unused at N = 4Mi, kept for generality).
__global__ void sph_tail_kernel(const float* __restrict__ in,
                                float* __restrict__ out,
                                long start, long npts) {
  long i = start + (long)blockIdx.x * blockDim.x + threadIdx.x;
  if (i >= npts) return;
  float vx = in[3 * i], vy = in[3 * i + 1], vz = in[3 * i + 2];
  float inv = 1.0f / fmaxf(sqrtf(vx * vx + vy * vy + vz * vz), 1e-12f);
  float x = vx * inv, y = vy * inv, z = vz * inv;
  float x2 = x * x, y2 = y * y, z2 = z * z;
  float* o = out + i * 16;
  o[0]  = 1.0f;
  o[1]  = S3 * y;
  o[2]  = S3 * z;
  o[3]  = S3 * x;
  o[4]  = S15 * x * y;
  o[5]  = S15 * y * z;
  o[6]  = 0.5f * S5 * (3.0f * z2 - 1.0f);
  o[7]  = S15 * x * z;
  o[8]  = 0.5f * S15 * (x2 - y2);
  o[9]  = 0.25f * S70 * y * (3.0f * x2 - y2);
  o[10] = S105 * x * y * z;
  o[11] = 0.25f * S42 * y * (5.0f * z2 - 1.0f);
  o[12] = 0.5f * S7 * z * (5.0f * z2 - 3.0f);
  o[13] = 0.25f * S42 * x * (5.0f * z2 - 1.0f);
  o[14] = 0.5f * S105 * z * (x2 - y2);
  o[15] = 0.25f * S70 * x * (x2 - 3.0f * y2);
}

extern "C" void kernel_launch(void* const* d_in, const int* in_sizes, int n_in,
                              void* d_out, int out_size, void* d_ws, size_t ws_size,
                              hipStream_t stream) {
  const float* in  = (const float*)d_in[0];
  float*       out = (float*)d_out;
  const long npts   = (long)in_sizes[0] / 3;
  const long nTiles = npts / 16;

  if (nTiles > 0) {
    const int  block        = 256;                 // 8 waves per block
    long blocksNeeded       = (nTiles + 7) / 8;
    const int  grid         = (int)(blocksNeeded > 8192 ? 8192 : blocksNeeded);
    sph_wmma_kernel<<<grid, block, 0, stream>>>(in, out, nTiles);
  }
  const long rem = npts - nTiles * 16;
  if (rem > 0) {
    const int block = 64;
    const int grid  = (int)((rem + block - 1) / block);
    sph_tail_kernel<<<grid, block, 0, stream>>>(in, out, nTiles * 16, npts);
  }
}